// RNN_37323265802941
// MI455X (gfx1250) — compile-verified
//
#include <hip/hip_runtime.h>
#include <math.h>

// Problem constants from the reference: T=20, S=8192, D=100, H=81, O=2
#define TT 20
#define SS 8192
#define DD 100
#define HH 81
#define OO 2
#define MROWS (TT * SS)          // 163840
#define W1STRIDE (DD + HH)       // 181

typedef float v2f __attribute__((ext_vector_type(2)));
typedef float v8f __attribute__((ext_vector_type(8)));

// ---------------------------------------------------------------------------
// Phase 1: x_proj[m][n] = sum_k x[m][k] * Wx[n][k] + b1[n]
// M=163840, K=100 (25 k-steps of 4), N=81 padded to 96 (6 tiles of 16).
// One wave per 16-row M-tile, covering all 6 N-tiles (6 f32 accumulators).
// Uses V_WMMA_F32_16X16X4_F32 (native f32 WMMA — no precision downcast).
// ---------------------------------------------------------------------------
__global__ void __launch_bounds__(256)
xproj_gemm_wmma(const float* __restrict__ x,     // (163840, 100)
                const float* __restrict__ W1,    // (81, 181), Wx = W1[:, :100]
                const float* __restrict__ b1,    // (81,)
                float* __restrict__ xproj)       // (163840, 81)
{
    __shared__ float lwx[96 * DD];   // Wx zero-padded to 96 rows, stride 100
    __shared__ float lb1[96];

    const int tid = threadIdx.x;
    for (int idx = tid; idx < 96 * DD; idx += 256) {
        int n = idx / DD, k = idx % DD;
        lwx[idx] = (n < HH) ? W1[n * W1STRIDE + k] : 0.0f;
    }
    if (tid < 96) lb1[tid] = (tid < HH) ? b1[tid] : 0.0f;
    __syncthreads();

    const int wave = tid >> 5;
    const int lane = tid & 31;
    const int hf   = lane >> 4;       // 0: K-pair {k,k+1}, 1: {k+2,k+3}
    const int l16  = lane & 15;
    const long mtile = (long)blockIdx.x * 8 + wave;
    const long m0    = mtile * 16;

    const float* arow = x + (m0 + l16) * DD + hf * 2;

    v8f acc[6];
#pragma unroll
    for (int nt = 0; nt < 6; ++nt)
#pragma unroll
        for (int r = 0; r < 8; ++r) acc[nt][r] = 0.0f;

#pragma unroll 5
    for (int k = 0; k < DD; k += 4) {
        // A frag: lane m<16 -> {x[m][k], x[m][k+1]}; lane m+16 -> {x[m][k+2], x[m][k+3]}
        v2f a = *(const v2f*)(arow + k);
#pragma unroll
        for (int nt = 0; nt < 6; ++nt) {
            // B frag: B[kk][n] = WxT[kk][n] = Wx[n][kk], mirrored K layout
            v2f b = *(const v2f*)&lwx[(nt * 16 + l16) * DD + k + hf * 2];
            acc[nt] = __builtin_amdgcn_wmma_f32_16x16x4_f32(
                false, a, false, b, (short)0, acc[nt], false, false);
        }
    }

    // C layout: VGPR r -> row m0 + r + hf*8, col = nt*16 + l16
#pragma unroll
    for (int nt = 0; nt < 6; ++nt) {
        int col = nt * 16 + l16;
        if (col < HH) {
            float bias = lb1[col];
#pragma unroll
            for (int r = 0; r < 8; ++r) {
                long row = m0 + r + hf * 8;
                xproj[row * HH + col] = acc[nt][r] + bias;
            }
        }
    }
}

// ---------------------------------------------------------------------------
// Phase 2: sequential scan h = tanh(xproj[i] + Wh @ h), 163840 steps.
// Single workgroup, 128 threads (thread j < 81 owns output row j).
// Wh (81x81, 26 KB) and h live in LDS. Stride-81 rows are conflict-free
// (gcd(81,64)=1); h[k] reads are uniform-address broadcasts.
// Emits user_vec rows at i % 8192 == 8191.
// ---------------------------------------------------------------------------
__global__ void __launch_bounds__(128)
rnn_scan(const float* __restrict__ xp,        // (163840, 81)
         const float* __restrict__ W1,        // Wh = W1[:, 100:181]
         const float* __restrict__ hidden0,   // (81,)
         float* __restrict__ user_vec)        // (20, 81)
{
    __shared__ float wh[HH * HH];
    __shared__ float h[96];

    const int tid = threadIdx.x;
    for (int idx = tid; idx < HH * HH; idx += 128) {
        int r = idx / HH, c = idx % HH;
        wh[idx] = W1[r * W1STRIDE + DD + c];
    }
    if (tid < HH) h[tid] = hidden0[tid];
    __syncthreads();

    const bool act = (tid < HH);
    const float* wr = &wh[tid * HH];

    for (long i = 0; i < (long)MROWS; ++i) {
        float acc = 0.0f;
        if (act) {
            acc = xp[i * HH + tid];
#pragma unroll 9
            for (int k = 0; k < HH; ++k)
                acc = fmaf(h[k], wr[k], acc);
            acc = tanhf(acc);
        }
        __syncthreads();                 // all reads of h done
        if (act) {
            h[tid] = acc;
            if ((i & (SS - 1)) == (SS - 1))
                user_vec[(i >> 13) * HH + tid] = acc;
        }
        __syncthreads();                 // new h visible
    }
}

// ---------------------------------------------------------------------------
// Phase 3: out[o] = sigmoid(user_vec . W2[o] + b2[o]), o in {0,1}
// ---------------------------------------------------------------------------
__global__ void __launch_bounds__(128)
head_kernel(const float* __restrict__ uv,     // (1620,)
            const float* __restrict__ W2,     // (2, 1620)
            const float* __restrict__ b2,     // (2,)
            float* __restrict__ out)          // (2,)
{
    __shared__ float red0[128];
    __shared__ float red1[128];
    const int tid = threadIdx.x;
    float s0 = 0.0f, s1 = 0.0f;
    for (int idx = tid; idx < TT * HH; idx += 128) {
        float u = uv[idx];
        s0 = fmaf(u, W2[idx], s0);
        s1 = fmaf(u, W2[TT * HH + idx], s1);
    }
    red0[tid] = s0;
    red1[tid] = s1;
    __syncthreads();
    for (int off = 64; off > 0; off >>= 1) {
        if (tid < off) {
            red0[tid] += red0[tid + off];
            red1[tid] += red1[tid + off];
        }
        __syncthreads();
    }
    if (tid == 0) {
        out[0] = 1.0f / (1.0f + expf(-(red0[0] + b2[0])));
        out[1] = 1.0f / (1.0f + expf(-(red1[0] + b2[1])));
    }
}

// ---------------------------------------------------------------------------
extern "C" void kernel_launch(void* const* d_in, const int* in_sizes, int n_in,
                              void* d_out, int out_size, void* d_ws, size_t ws_size,
                              hipStream_t stream)
{
    const float* input  = (const float*)d_in[0];   // (20, 8192, 100)
    const float* hidden = (const float*)d_in[1];   // (81,)
    const float* W1     = (const float*)d_in[2];   // (81, 181)
    const float* b1     = (const float*)d_in[3];   // (81,)
    const float* W2     = (const float*)d_in[4];   // (2, 1620)
    const float* b2     = (const float*)d_in[5];   // (2,)
    float* out = (float*)d_out;

    float* xproj = (float*)d_ws;                           // 163840*81 floats (~53 MB)
    float* uv    = xproj + (size_t)MROWS * HH;             // 1620 floats

    // 10240 M-tiles of 16 rows, 8 waves (tiles) per 256-thread block
    xproj_gemm_wmma<<<MROWS / 128, 256, 0, stream>>>(input, W1, b1, xproj);
    rnn_scan<<<1, 128, 0, stream>>>(xproj, W1, hidden, uv);
    head_kernel<<<1, 128, 0, stream>>>(uv, W2, b2, out);
}